// MDLoss_61942018343087
// MI455X (gfx1250) — compile-verified
//
#include <hip/hip_runtime.h>

typedef __attribute__((ext_vector_type(2))) float v2f;
typedef __attribute__((ext_vector_type(8))) float v8f;

#define BATCH 256
#define NP    512
#define NG    1024

// One block: 8 waves, each wave owns a 16-wide p-tile; block covers 128 p's of one batch.
// Grid: 256 batches * 4 blocks = 1024 blocks.
__global__ __launch_bounds__(256) void mdloss_main_kernel(
    const float* __restrict__ ini,        // [B, NP, 2]
    const float* __restrict__ pred,       // [B, NP, 2]
    const float* __restrict__ gt,         // [B, NG, 2]
    float* __restrict__ block_sums)       // [gridDim.x]
{
    __shared__ float lx[NG];   // x_g
    __shared__ float ly[NG];   // y_g
    __shared__ float lq[NG];   // x_g^2 + y_g^2
    __shared__ float wsum[8];

    const int b    = blockIdx.x >> 2;
    const int pblk = (blockIdx.x & 3) * 128;
    const int tid  = threadIdx.x;
    const int wave = tid >> 5;
    const int lane = tid & 31;
    const int ln16 = lane & 15;
    const bool hi  = (lane >= 16);

    // Stage GT points for this batch into LDS, precompute |g|^2.
    const float* gtb = gt + (size_t)b * NG * 2;
    for (int g = tid; g < NG; g += 256) {
        float x = gtb[g * 2 + 0];
        float y = gtb[g * 2 + 1];
        lx[g] = x;
        ly[g] = y;
        lq[g] = __builtin_fmaf(x, x, y * y);
    }
    __syncthreads();

    // B fragment (K x N = 4x16): lanes 0-15 hold K=0,1 ; lanes 16-31 hold K=2,3.
    // Column p = (-2*ix, -2*iy, 1, 0).
    const int p = pblk + wave * 16 + ln16;
    const float* inib = ini + ((size_t)b * NP + p) * 2;
    const float ix = inib[0];
    const float iy = inib[1];
    v2f bfrag;
    bfrag.x = hi ? 1.0f : (-2.0f * ix);
    bfrag.y = hi ? 0.0f : (-2.0f * iy);

    // A fragment (M x K = 16x4): lanes 0-15 hold K=0,1 (x_g, y_g);
    // lanes 16-31 hold K=2,3 (|g|^2, 0). Branchless via per-lane LDS base.
    const float* a0base = hi ? lq : lx;

    float best  = 3.4e38f;
    int   bestg = 0;

    for (int t = 0; t < NG / 16; ++t) {
        const int g0 = t * 16;
        v2f afrag;
        afrag.x = a0base[g0 + ln16];
        float yv = ly[g0 + ln16];
        afrag.y = hi ? 0.0f : yv;

        v8f c = {};
        // score[g,p] = |g|^2 - 2 g.a  (argmin-equivalent to squared distance)
        c = __builtin_amdgcn_wmma_f32_16x16x4_f32(
                /*neg_a=*/false, afrag, /*neg_b=*/false, bfrag,
                /*c_mod=*/(short)0, c, /*reuse_a=*/false, /*reuse_b=*/false);

        // D layout: VGPR r, lanes 0-15 -> M=r ; lanes 16-31 -> M=r+8. N = lane%16 = p.
        const int gb = g0 + (hi ? 8 : 0);
#pragma unroll
        for (int r = 0; r < 8; ++r) {
            float s  = c[r];
            int   gc = gb + r;
            bool lt  = (s < best);
            best  = lt ? s  : best;
            bestg = lt ? gc : bestg;
        }
    }

    // Merge the two lane-halves (they hold disjoint g rows of the same p column).
    {
        float ob = __shfl_xor(best, 16, 32);
        int   og = __shfl_xor(bestg, 16, 32);
        if (ob < best || (ob == best && og < bestg)) { best = ob; bestg = og; }
    }

    // Lanes 0-15 own unique p's: compute L1 loss vs matched GT point.
    float loss = 0.0f;
    if (!hi) {
        const float* predb = pred + ((size_t)b * NP + p) * 2;
        float gx = lx[bestg];
        float gy = ly[bestg];
        loss = fabsf(predb[0] - gx) + fabsf(predb[1] - gy);
    }

    // Deterministic wave reduction (hi lanes contribute 0).
#pragma unroll
    for (int m = 16; m >= 1; m >>= 1) loss += __shfl_xor(loss, m, 32);
    if (lane == 0) wsum[wave] = loss;
    __syncthreads();
    if (tid == 0) {
        float s = 0.0f;
#pragma unroll
        for (int w = 0; w < 8; ++w) s += wsum[w];
        block_sums[blockIdx.x] = s;
    }
}

__global__ __launch_bounds__(256) void mdloss_reduce_kernel(
    const float* __restrict__ block_sums, float* __restrict__ out, int n)
{
    __shared__ float wsum[8];
    const int tid = threadIdx.x;
    float s = 0.0f;
    for (int i = tid; i < n; i += 256) s += block_sums[i];
#pragma unroll
    for (int m = 16; m >= 1; m >>= 1) s += __shfl_xor(s, m, 32);
    const int lane = tid & 31;
    const int wave = tid >> 5;
    if (lane == 0) wsum[wave] = s;
    __syncthreads();
    if (tid == 0) {
        float t = 0.0f;
#pragma unroll
        for (int w = 0; w < 8; ++w) t += wsum[w];
        out[0] = t * (1.0f / ((float)BATCH * (float)NP * 2.0f));
    }
}

extern "C" void kernel_launch(void* const* d_in, const int* in_sizes, int n_in,
                              void* d_out, int out_size, void* d_ws, size_t ws_size,
                              hipStream_t stream) {
    const float* ini  = (const float*)d_in[0];  // ini_pred_poly [256,512,2]
    const float* pred = (const float*)d_in[1];  // pred_polys_   [256,512,2]
    const float* gt   = (const float*)d_in[2];  // gt_polys      [256,1024,2]
    float* out = (float*)d_out;
    float* block_sums = (float*)d_ws;           // 1024 floats

    const int nblocks = BATCH * 4;              // 1024
    mdloss_main_kernel<<<nblocks, 256, 0, stream>>>(ini, pred, gt, block_sums);
    mdloss_reduce_kernel<<<1, 256, 0, stream>>>(block_sums, out, nblocks);
    (void)in_sizes; (void)n_in; (void)out_size; (void)ws_size;
}